// MambaBlock_52278341927350
// MI455X (gfx1250) — compile-verified
//
#include <hip/hip_runtime.h>
#include <math.h>

typedef float v2f __attribute__((ext_vector_type(2)));
typedef float v8f __attribute__((ext_vector_type(8)));

#define DIM      1024
#define D_STATE  16
#define D_CONV   4
#define D_INNER  2048
#define DT_RANK  64
#define B_SZ     2
#define L_SZ     2048
#define ROWS     (B_SZ * L_SZ)          // 4096

#define LDSTR    20                     // LDS row stride: float4-aligned + conflict-free

// ---------------------------------------------------------------------------
// LayerNorm: one block per row of x [ROWS, DIM]
// ---------------------------------------------------------------------------
__global__ __launch_bounds__(256) void ln_kernel(const float* __restrict__ x,
                                                 const float* __restrict__ gamma,
                                                 const float* __restrict__ beta,
                                                 float* __restrict__ xn) {
    const int row = blockIdx.x;
    const float* xr = x + (size_t)row * DIM;
    float s = 0.f, s2 = 0.f;
    for (int i = threadIdx.x; i < DIM; i += 256) {
        float v = xr[i];
        s += v; s2 += v * v;
    }
    __shared__ float r0[256], r1[256];
    r0[threadIdx.x] = s; r1[threadIdx.x] = s2;
    __syncthreads();
    for (int off = 128; off > 0; off >>= 1) {
        if (threadIdx.x < off) {
            r0[threadIdx.x] += r0[threadIdx.x + off];
            r1[threadIdx.x] += r1[threadIdx.x + off];
        }
        __syncthreads();
    }
    const float mu  = r0[0] * (1.f / DIM);
    const float var = r1[0] * (1.f / DIM) - mu * mu;
    const float rstd = rsqrtf(var + 1e-5f);
    float* outr = xn + (size_t)row * DIM;
    for (int i = threadIdx.x; i < DIM; i += 256)
        outr[i] = (xr[i] - mu) * rstd * gamma[i] + beta[i];
}

// ---------------------------------------------------------------------------
// Generic fp32 WMMA GEMM: C[M,N] = act(A[M,K(lda)] * B[N,K]^T + bias) (+Res)
// block = 256 threads (8 waves), 128x64 C tile, wave computes 32x32 (2x2
// accumulators), K chunked by 16 (4 x k=4 WMMA steps).
// mode: 0 = none, 1 = softplus, 2 = add residual
// ---------------------------------------------------------------------------
__global__ __launch_bounds__(256) void gemm_wmma_f32(const float* __restrict__ A, int lda,
                                                     const float* __restrict__ Bm, // [N][K]
                                                     float* __restrict__ C,
                                                     const float* __restrict__ bias,
                                                     const float* __restrict__ Res,
                                                     int M, int N, int K, int mode) {
    __shared__ float As[128 * LDSTR];
    __shared__ float Bs[64 * LDSTR];

    const int t    = threadIdx.x;
    const int wave = t >> 5;
    const int lane = t & 31;
    const int half = lane >> 4;
    const int lm   = lane & 15;

    const int m0   = blockIdx.y * 128;
    const int n0   = blockIdx.x * 64;
    const int mrow = (wave & 3) * 32;   // wave's m offset inside block tile
    const int ncol = (wave >> 2) * 32;  // wave's n offset

    const int lrow = t >> 2;            // 0..63: row loaded by this thread
    const int lk   = (t & 3) * 4;       // 0,4,8,12: k quad

    v8f acc00 = {}, acc01 = {}, acc10 = {}, acc11 = {};

    for (int kc = 0; kc < K; kc += 16) {
        // --- cooperative stage A (128 x 16) : 2 float4 per thread ---
        const float4 av0 = *(const float4*)(A + (size_t)(m0 + lrow)      * lda + kc + lk);
        const float4 av1 = *(const float4*)(A + (size_t)(m0 + lrow + 64) * lda + kc + lk);
        *(float4*)&As[ lrow       * LDSTR + lk] = av0;
        *(float4*)&As[(lrow + 64) * LDSTR + lk] = av1;
        // --- cooperative stage B (64 n x 16 k), N-guarded ---
        float4 bv = make_float4(0.f, 0.f, 0.f, 0.f);
        const int gn = n0 + lrow;
        if (gn < N) bv = *(const float4*)(Bm + (size_t)gn * K + kc + lk);
        *(float4*)&Bs[lrow * LDSTR + lk] = bv;

        // prefetch next chunk while we compute (global_prefetch_b8)
        if (kc + 16 < K) {
            __builtin_prefetch(A + (size_t)(m0 + lrow)      * lda + kc + 16 + lk, 0, 3);
            __builtin_prefetch(A + (size_t)(m0 + lrow + 64) * lda + kc + 16 + lk, 0, 3);
            if (gn < N) __builtin_prefetch(Bm + (size_t)gn * K + kc + 16 + lk, 0, 3);
        }
        __syncthreads();

#pragma unroll
        for (int j = 0; j < 4; ++j) {
            const int k2 = j * 4 + half * 2;   // ISA A/B lane layout for 16x16x4 f32
            v2f a0, a1, b0, b1;
            a0.x = As[(mrow + lm)      * LDSTR + k2];
            a0.y = As[(mrow + lm)      * LDSTR + k2 + 1];
            a1.x = As[(mrow + 16 + lm) * LDSTR + k2];
            a1.y = As[(mrow + 16 + lm) * LDSTR + k2 + 1];
            b0.x = Bs[(ncol + lm)      * LDSTR + k2];
            b0.y = Bs[(ncol + lm)      * LDSTR + k2 + 1];
            b1.x = Bs[(ncol + 16 + lm) * LDSTR + k2];
            b1.y = Bs[(ncol + 16 + lm) * LDSTR + k2 + 1];
            acc00 = __builtin_amdgcn_wmma_f32_16x16x4_f32(false, a0, false, b0,
                                                          (short)0, acc00, false, false);
            acc01 = __builtin_amdgcn_wmma_f32_16x16x4_f32(false, a0, false, b1,
                                                          (short)0, acc01, false, false);
            acc10 = __builtin_amdgcn_wmma_f32_16x16x4_f32(false, a1, false, b0,
                                                          (short)0, acc10, false, false);
            acc11 = __builtin_amdgcn_wmma_f32_16x16x4_f32(false, a1, false, b1,
                                                          (short)0, acc11, false, false);
        }
        __syncthreads();
    }

    // --- epilogue: VGPR r holds row (r + 8*half) of each 16x16 tile ---
#pragma unroll
    for (int r = 0; r < 8; ++r) {
#pragma unroll
        for (int mi = 0; mi < 2; ++mi) {
            const int gm = m0 + mrow + mi * 16 + half * 8 + r;
#pragma unroll
            for (int nj = 0; nj < 2; ++nj) {
                const int gn = n0 + ncol + nj * 16 + lm;
                if (gn >= N) continue;
                float v;
                if (mi == 0) v = (nj == 0) ? acc00[r] : acc01[r];
                else         v = (nj == 0) ? acc10[r] : acc11[r];
                if (bias) v += bias[gn];
                if (mode == 1) v = (v > 20.f) ? v : log1pf(__expf(v));
                if (mode == 2) v += Res[(size_t)gm * N + gn];
                C[(size_t)gm * N + gn] = v;
            }
        }
    }
}

// ---------------------------------------------------------------------------
// Causal depthwise conv (k=4) over seq + SiLU. u part = xz[:, 0:2048].
// ---------------------------------------------------------------------------
__global__ __launch_bounds__(256) void conv_silu_kernel(const float* __restrict__ xz,
                                                        const float* __restrict__ cw,
                                                        const float* __restrict__ cb,
                                                        float* __restrict__ u) {
    const size_t idx = (size_t)blockIdx.x * 256 + threadIdx.x; // ROWS*D_INNER
    const int c  = (int)(idx & (D_INNER - 1));
    const size_t bl = idx >> 11;           // 0..ROWS-1  (= b*L + l)
    const int l  = (int)(bl & (L_SZ - 1));
    float acc = cb[c];
#pragma unroll
    for (int k = 0; k < D_CONV; ++k) {
        const int ls = l - (D_CONV - 1) + k;
        if (ls >= 0)
            acc += cw[c * D_CONV + k] * xz[(bl + (size_t)(ls - l)) * (2 * D_INNER) + c];
    }
    u[bl * D_INNER + c] = acc / (1.f + __expf(-acc)); // silu
}

// ---------------------------------------------------------------------------
// Selective scan. One lane per (d, n) state; wave32 = 2 channels x 16 states.
// Fuses D-skip and SiLU(z) gate; writes gated y in place over u (safe: each
// half-wave exclusively owns its channel d, reads u before the lane-0 store).
// ---------------------------------------------------------------------------
__global__ __launch_bounds__(256) void scan_kernel(const float* __restrict__ delta,
                                                   float* __restrict__ u,      // in: silu(conv), out: gated y
                                                   const float* __restrict__ xdbl,
                                                   const float* __restrict__ xz,
                                                   const float* __restrict__ A_log,
                                                   const float* __restrict__ Dp) {
    const int lane = threadIdx.x & 31;
    const int n    = lane & 15;
    const int ds   = lane >> 4;
    const int waveId = (int)((blockIdx.x * 256 + threadIdx.x) >> 5); // 0 .. B*D_INNER/2-1
    const int b  = waveId / (D_INNER / 2);
    const int dp = waveId % (D_INNER / 2);
    const int d  = dp * 2 + ds;

    const float a  = -__expf(A_log[d * D_STATE + n]);
    const float Dd = Dp[d];
    float h = 0.f;
    const size_t rowBase = (size_t)b * L_SZ;

    for (int tt = 0; tt < L_SZ; ++tt) {
        const size_t r = rowBase + tt;
        const float dt = delta[r * D_INNER + d];
        const float uu = u[r * D_INNER + d];
        const float Bv = xdbl[r * (DT_RANK + 2 * D_STATE) + DT_RANK + n];
        const float Cv = xdbl[r * (DT_RANK + 2 * D_STATE) + DT_RANK + D_STATE + n];
        h = __expf(dt * a) * h + (dt * uu) * Bv;
        float y = h * Cv;
        y += __shfl_xor(y, 1, 32);
        y += __shfl_xor(y, 2, 32);
        y += __shfl_xor(y, 4, 32);
        y += __shfl_xor(y, 8, 32);
        if (n == 0) {
            const float zv = xz[r * (2 * D_INNER) + D_INNER + d];
            const float g  = zv / (1.f + __expf(-zv));
            u[r * D_INNER + d] = (y + uu * Dd) * g;
        }
    }
}

// ---------------------------------------------------------------------------
extern "C" void kernel_launch(void* const* d_in, const int* in_sizes, int n_in,
                              void* d_out, int out_size, void* d_ws, size_t ws_size,
                              hipStream_t stream) {
    const float* x      = (const float*)d_in[0];
    const float* gamma  = (const float*)d_in[1];
    const float* beta   = (const float*)d_in[2];
    const float* W_in   = (const float*)d_in[3];   // [4096, 1024]
    const float* conv_w = (const float*)d_in[4];   // [2048, 4]
    const float* conv_b = (const float*)d_in[5];   // [2048]
    const float* W_x    = (const float*)d_in[6];   // [96, 2048]
    const float* W_dt   = (const float*)d_in[7];   // [2048, 64]
    const float* b_dt   = (const float*)d_in[8];   // [2048]
    const float* A_log  = (const float*)d_in[9];   // [2048, 16]
    const float* Dp     = (const float*)d_in[10];  // [2048]
    const float* W_out  = (const float*)d_in[11];  // [1024, 2048]
    float* out = (float*)d_out;

    // workspace layout (floats)
    float* ws   = (float*)d_ws;
    float* xn   = ws;                                    // ROWS * DIM
    float* xz   = xn   + (size_t)ROWS * DIM;             // ROWS * 4096
    float* u    = xz   + (size_t)ROWS * 2 * D_INNER;     // ROWS * 2048 (also ygated)
    float* xdbl = u    + (size_t)ROWS * D_INNER;         // ROWS * 96
    float* dl   = xdbl + (size_t)ROWS * (DT_RANK + 2 * D_STATE); // ROWS * 2048

    // 1) LayerNorm
    ln_kernel<<<ROWS, 256, 0, stream>>>(x, gamma, beta, xn);

    // 2) in_proj: xz = xn @ W_in^T   [4096 x 4096]
    gemm_wmma_f32<<<dim3(2 * D_INNER / 64, ROWS / 128), 256, 0, stream>>>(
        xn, DIM, W_in, xz, nullptr, nullptr, ROWS, 2 * D_INNER, DIM, 0);

    // 3) causal depthwise conv + SiLU -> u
    conv_silu_kernel<<<(ROWS * D_INNER) / 256, 256, 0, stream>>>(xz, conv_w, conv_b, u);

    // 4) x_proj: xdbl = u @ W_x^T   [4096 x 96]
    gemm_wmma_f32<<<dim3(2, ROWS / 128), 256, 0, stream>>>(
        u, D_INNER, W_x, xdbl, nullptr, nullptr, ROWS, DT_RANK + 2 * D_STATE, D_INNER, 0);

    // 5) delta = softplus(dt @ W_dt^T + b_dt); dt = xdbl[:, :64] (lda = 96)
    gemm_wmma_f32<<<dim3(D_INNER / 64, ROWS / 128), 256, 0, stream>>>(
        xdbl, DT_RANK + 2 * D_STATE, W_dt, dl, b_dt, nullptr, ROWS, D_INNER, DT_RANK, 1);

    // 6) selective scan + D-skip + gate (in place over u)
    scan_kernel<<<(B_SZ * D_INNER / 2) / 8, 256, 0, stream>>>(dl, u, xdbl, xz, A_log, Dp);

    // 7) out_proj + residual: out = ygated @ W_out^T + x
    gemm_wmma_f32<<<dim3(DIM / 64, ROWS / 128), 256, 0, stream>>>(
        u, D_INNER, W_out, out, nullptr, x, ROWS, DIM, D_INNER, 2);
}